// AutoEncoder_86483461472567
// MI455X (gfx1250) — compile-verified
//
#include <hip/hip_runtime.h>
#include <hip/hip_bf16.h>

typedef __attribute__((ext_vector_type(2))) float v2f;
typedef __attribute__((ext_vector_type(8))) float v8f;

#define KT   32
#define KTP  36           // padded LDS stride: 144B rows -> 16B-aligned b128 async
                          // stores, and conflict-free 16-lane fragment reads
#define LRELU_SLOPE 0.01f

// Async global->LDS copy (CDNA5): per-lane 16B, tracked by ASYNCcnt.
__device__ __forceinline__ void async_b128(const float* g, unsigned lds_off) {
  asm volatile("global_load_async_to_lds_b128 %0, %1, off"
               :: "v"(lds_off), "v"((unsigned long long)(size_t)g)
               : "memory");
}
#define WAIT_ASYNC(n) asm volatile("s_wait_asynccnt " #n ::: "memory")

// =====================================================================
// Y = leaky_relu(X @ W^T + b)
// X:[M,K], W:[N,K] row-major, Y:[M,N]
// Block tile 64(M) x 128(N); 8 waves (2x4); wave tile 32x32.
// Double-buffered LDS fed by global_load_async_to_lds_b128.
// =====================================================================
__device__ __forceinline__ void gemm_issue(
    const float* __restrict__ X, const float* __restrict__ W, int K,
    int bm, int bn, int kt, float (*Xb)[KTP], float (*Wb)[KTP], int tid)
{
  #pragma unroll
  for (int it = 0; it < 2; ++it) {            // 64x32 floats = 512 float4
    int idx = tid + it * 256;
    int row = idx >> 3, c4 = (idx & 7) << 2;
    async_b128(X + (size_t)(bm + row) * K + kt + c4,
               (unsigned)(size_t)&Xb[row][c4]);
  }
  #pragma unroll
  for (int it = 0; it < 4; ++it) {            // 128x32 floats = 1024 float4
    int idx = tid + it * 256;
    int row = idx >> 3, c4 = (idx & 7) << 2;
    async_b128(W + (size_t)(bn + row) * K + kt + c4,
               (unsigned)(size_t)&Wb[row][c4]);
  }
}

__global__ __launch_bounds__(256) void gemm_lrelu_kernel(
    const float* __restrict__ X, const float* __restrict__ W,
    const float* __restrict__ bias, float* __restrict__ Y,
    int M, int N, int K)
{
  __shared__ alignas(16) float Xs[2][64][KTP];
  __shared__ alignas(16) float Ws[2][128][KTP];

  const int tid   = threadIdx.x;
  const int lane  = tid & 31;
  const int wid   = tid >> 5;
  const int wi    = wid >> 2;      // 0..1 (rows)
  const int wj    = wid & 3;       // 0..3 (cols)
  const int l16   = lane & 15;
  const int lhalf = lane >> 4;
  const int bm = blockIdx.y * 64;
  const int bn = blockIdx.x * 128;

  v8f acc[2][2];
  #pragma unroll
  for (int m = 0; m < 2; ++m)
    #pragma unroll
    for (int n = 0; n < 2; ++n)
      #pragma unroll
      for (int r = 0; r < 8; ++r) acc[m][n][r] = 0.0f;

  const int T = K / KT;
  gemm_issue(X, W, K, bm, bn, 0, Xs[0], Ws[0], tid);     // prologue: tile 0

  for (int t = 0; t < T; ++t) {
    const int cur = t & 1;
    if (t + 1 < T) {
      gemm_issue(X, W, K, bm, bn, (t + 1) * KT, Xs[cur ^ 1], Ws[cur ^ 1], tid);
      WAIT_ASYNC(0x6);            // 6 newer ops outstanding -> tile t complete
    } else {
      WAIT_ASYNC(0x0);
    }
    __syncthreads();              // tile t visible to all waves

    float (*Xc)[KTP] = Xs[cur];
    float (*Wc)[KTP] = Ws[cur];
    #pragma unroll
    for (int kk = 0; kk < KT; kk += 4) {
      const int kq = kk + lhalf * 2;   // even -> 8B-aligned ds_load_b64
      v2f a[2], b[2];
      #pragma unroll
      for (int m = 0; m < 2; ++m) a[m] = *(const v2f*)(&Xc[wi*32 + m*16 + l16][kq]);
      #pragma unroll
      for (int n = 0; n < 2; ++n) b[n] = *(const v2f*)(&Wc[wj*32 + n*16 + l16][kq]);
      #pragma unroll
      for (int m = 0; m < 2; ++m)
        #pragma unroll
        for (int n = 0; n < 2; ++n)
          acc[m][n] = __builtin_amdgcn_wmma_f32_16x16x4_f32(
              false, a[m], false, b[n], (short)0, acc[m][n], false, false);
    }
    __syncthreads();              // reads done before buffer is overwritten
  }

  #pragma unroll
  for (int n = 0; n < 2; ++n) {
    const int col = bn + wj*32 + n*16 + l16;
    const float bv = bias[col];
    #pragma unroll
    for (int m = 0; m < 2; ++m) {
      const int rowb = bm + wi*32 + m*16 + lhalf*8;
      #pragma unroll
      for (int r = 0; r < 8; ++r) {
        float y = acc[m][n][r] + bv;
        y = (y >= 0.0f) ? y : LRELU_SLOPE * y;
        Y[(size_t)(rowb + r) * N + col] = y;
      }
    }
  }
}

// =====================================================================
// rx[row] = sum_k src[row][k]^2
// =====================================================================
__global__ __launch_bounds__(256) void row_sumsq_kernel(
    const float* __restrict__ src, float* __restrict__ out, int K)
{
  const int row = blockIdx.x;
  const float* p = src + (size_t)row * K;
  float s = 0.0f;
  for (int k = threadIdx.x; k < K; k += 256) { float v = p[k]; s += v * v; }
  __shared__ float red[256];
  red[threadIdx.x] = s;
  __syncthreads();
  for (int off = 128; off > 0; off >>= 1) {
    if (threadIdx.x < off) red[threadIdx.x] += red[threadIdx.x + off];
    __syncthreads();
  }
  if (threadIdx.x == 0) out[row] = red[0];
}

// =====================================================================
// SYRK tile: acc[m][n] = src[ri..ri+128) . src[rj..rj+128)^T
// Block tile 128x128; 8 waves (4 rows x 2 cols); wave tile 32x64.
// Double-buffered async LDS pipeline (8 async ops per tile).
// =====================================================================
__device__ __forceinline__ void syrk_issue(
    const float* __restrict__ src, int K, int ri, int rj, int kt,
    float (*Ti)[KTP], float (*Tj)[KTP], int tid)
{
  #pragma unroll
  for (int it = 0; it < 4; ++it) {            // 128x32 floats = 1024 float4
    int idx = tid + it * 256;
    int row = idx >> 3, c4 = (idx & 7) << 2;
    async_b128(src + (size_t)(ri + row) * K + kt + c4,
               (unsigned)(size_t)&Ti[row][c4]);
    async_b128(src + (size_t)(rj + row) * K + kt + c4,
               (unsigned)(size_t)&Tj[row][c4]);
  }
}

__device__ __forceinline__ void syrk_tile(
    const float* __restrict__ src, int K, int ri, int rj,
    float (*Tib)[128][KTP], float (*Tjb)[128][KTP],
    v8f (&acc)[2][4],
    int tid, int wi, int wj, int l16, int lhalf)
{
  #pragma unroll
  for (int m = 0; m < 2; ++m)
    #pragma unroll
    for (int n = 0; n < 4; ++n)
      #pragma unroll
      for (int r = 0; r < 8; ++r) acc[m][n][r] = 0.0f;

  const int T = K / KT;
  syrk_issue(src, K, ri, rj, 0, Tib[0], Tjb[0], tid);

  for (int t = 0; t < T; ++t) {
    const int cur = t & 1;
    if (t + 1 < T) {
      syrk_issue(src, K, ri, rj, (t + 1) * KT, Tib[cur ^ 1], Tjb[cur ^ 1], tid);
      WAIT_ASYNC(0x8);
    } else {
      WAIT_ASYNC(0x0);
    }
    __syncthreads();

    float (*Ti)[KTP] = Tib[cur];
    float (*Tj)[KTP] = Tjb[cur];
    #pragma unroll
    for (int kk = 0; kk < KT; kk += 4) {
      const int kq = kk + lhalf * 2;
      v2f a[2], b[4];
      #pragma unroll
      for (int m = 0; m < 2; ++m) a[m] = *(const v2f*)(&Ti[wi*32 + m*16 + l16][kq]);
      #pragma unroll
      for (int n = 0; n < 4; ++n) b[n] = *(const v2f*)(&Tj[wj*64 + n*16 + l16][kq]);
      #pragma unroll
      for (int m = 0; m < 2; ++m)
        #pragma unroll
        for (int n = 0; n < 4; ++n)
          acc[m][n] = __builtin_amdgcn_wmma_f32_16x16x4_f32(
              false, a[m], false, b[n], (short)0, acc[m][n], false, false);
    }
    __syncthreads();
  }
}

// =====================================================================
// Fused distance + statistics over upper-triangle 128x128 tiles.
// G1 = x.x^T (K=1024), G2 = lat.lat^T (K=128) via WMMA; d1,d2 formed in
// registers; 5 sums accumulated in f64 (off-diagonal tiles weight 2).
// Every block writes its partials -> no stale state, deterministic.
// =====================================================================
__global__ __launch_bounds__(256) void dist_stats_kernel(
    const float* __restrict__ X, const float* __restrict__ L,
    const float* __restrict__ rx, const float* __restrict__ rl,
    double* __restrict__ partials)
{
  const int bi = blockIdx.y, bj = blockIdx.x;
  const int bid = bi * gridDim.x + bj;
  const int tid = threadIdx.x;
  if (bj < bi) {                                  // lower triangle
    if (tid < 5) partials[(size_t)bid * 5 + tid] = 0.0;
    return;
  }

  __shared__ alignas(16) float Ti[2][128][KTP];
  __shared__ alignas(16) float Tj[2][128][KTP];
  __shared__ double sred[5][256];

  const int lane  = tid & 31;
  const int wid   = tid >> 5;
  const int wi    = wid >> 1;      // 0..3 (row groups of 32)
  const int wj    = wid & 1;       // 0..1 (col groups of 64)
  const int l16   = lane & 15;
  const int lhalf = lane >> 4;

  v8f acc1[2][4], acc2[2][4];
  syrk_tile(X, 1024, bi * 128, bj * 128, Ti, Tj, acc1, tid, wi, wj, l16, lhalf);
  syrk_tile(L,  128, bi * 128, bj * 128, Ti, Tj, acc2, tid, wi, wj, l16, lhalf);

  float rxj[4], rlj[4];
  #pragma unroll
  for (int n = 0; n < 4; ++n) {
    const int j = bj*128 + wj*64 + n*16 + l16;
    rxj[n] = rx[j]; rlj[n] = rl[j];
  }

  double s1 = 0.0, s2 = 0.0, q1 = 0.0, q2 = 0.0, pp = 0.0;
  #pragma unroll
  for (int m = 0; m < 2; ++m) {
    #pragma unroll
    for (int r = 0; r < 8; ++r) {
      const int i = bi*128 + wi*32 + m*16 + lhalf*8 + r;
      const float rxi = rx[i], rli = rl[i];
      #pragma unroll
      for (int n = 0; n < 4; ++n) {
        float sq1 = fmaxf(rxi + rxj[n] - 2.0f * acc1[m][n][r], 0.0f);
        float sq2 = fmaxf(rli + rlj[n] - 2.0f * acc2[m][n][r], 0.0f);
        float d1 = sqrtf(sq1);
        float d2 = sqrtf(sq2);
        s1 += (double)d1;  s2 += (double)d2;
        q1 += (double)sq1; q2 += (double)sq2;
        pp += (double)d1 * (double)d2;
      }
    }
  }

  const double w = (bi == bj) ? 1.0 : 2.0;        // symmetry weight
  sred[0][tid] = s1; sred[1][tid] = s2; sred[2][tid] = q1; sred[3][tid] = q2; sred[4][tid] = pp;
  __syncthreads();
  for (int off = 128; off > 0; off >>= 1) {
    if (tid < off) {
      #pragma unroll
      for (int c = 0; c < 5; ++c) sred[c][tid] += sred[c][tid + off];
    }
    __syncthreads();
  }
  if (tid < 5) partials[(size_t)bid * 5 + tid] = sred[tid][0] * w;
}

// =====================================================================
// Deterministic final reduction -> corr scalar.
// corr = (P/n - m1*m2) / sqrt(v1*v2), v = (Q - n*m^2)/(n-1)
// =====================================================================
__global__ __launch_bounds__(256) void reduce_corr_kernel(
    const double* __restrict__ partials, int nblocks, float* __restrict__ out)
{
  __shared__ double red[5][256];
  const int tid = threadIdx.x;
  double s[5] = {0.0, 0.0, 0.0, 0.0, 0.0};
  for (int b = tid; b < nblocks; b += 256) {
    #pragma unroll
    for (int c = 0; c < 5; ++c) s[c] += partials[(size_t)b * 5 + c];
  }
  #pragma unroll
  for (int c = 0; c < 5; ++c) red[c][tid] = s[c];
  __syncthreads();
  for (int off = 128; off > 0; off >>= 1) {
    if (tid < off) {
      #pragma unroll
      for (int c = 0; c < 5; ++c) red[c][tid] += red[c][tid + off];
    }
    __syncthreads();
  }
  if (tid == 0) {
    const double n = 4096.0 * 4096.0;
    const double S1 = red[0][0], S2 = red[1][0];
    const double Q1 = red[2][0], Q2 = red[3][0], P = red[4][0];
    const double m1 = S1 / n, m2 = S2 / n;
    const double v1 = (Q1 - n * m1 * m1) / (n - 1.0);
    const double v2 = (Q2 - n * m2 * m2) / (n - 1.0);
    const double cov = P / n - m1 * m2;
    out[0] = (float)(cov / sqrt(v1 * v2));
  }
}

// =====================================================================
extern "C" void kernel_launch(void* const* d_in, const int* in_sizes, int n_in,
                              void* d_out, int out_size, void* d_ws, size_t ws_size,
                              hipStream_t stream)
{
  (void)in_sizes; (void)n_in; (void)out_size; (void)ws_size;
  const float* x   = (const float*)d_in[0];
  const float* We1 = (const float*)d_in[1];
  const float* be1 = (const float*)d_in[2];
  const float* We2 = (const float*)d_in[3];
  const float* be2 = (const float*)d_in[4];
  const float* We3 = (const float*)d_in[5];
  const float* be3 = (const float*)d_in[6];
  const float* Wd1 = (const float*)d_in[7];
  const float* bd1 = (const float*)d_in[8];
  const float* Wd2 = (const float*)d_in[9];
  const float* bd2 = (const float*)d_in[10];
  const float* Wd3 = (const float*)d_in[11];
  const float* bd3 = (const float*)d_in[12];

  float* pred = (float*)d_out;                       // [4096,1024]
  float* corr = pred + (size_t)4096 * 1024;          // scalar

  char* ws = (char*)d_ws;
  double* partials = (double*)ws;                    // 1024 blocks * 5 doubles
  float* h1  = (float*)(ws + (64 << 10));            // [4096,512]
  float* h2  = h1  + (size_t)4096 * 512;             // [4096,256]
  float* lat = h2  + (size_t)4096 * 256;             // [4096,128]
  float* dd1 = lat + (size_t)4096 * 128;             // [4096,256]
  float* dd2 = dd1 + (size_t)4096 * 256;             // [4096,512]
  float* rxv = dd2 + (size_t)4096 * 512;             // [4096]
  float* rlv = rxv + 4096;                           // [4096]

  const dim3 blk(256);
  gemm_lrelu_kernel<<<dim3(4, 64), blk, 0, stream>>>(x,   We1, be1, h1,   4096,  512, 1024);
  gemm_lrelu_kernel<<<dim3(2, 64), blk, 0, stream>>>(h1,  We2, be2, h2,   4096,  256,  512);
  gemm_lrelu_kernel<<<dim3(1, 64), blk, 0, stream>>>(h2,  We3, be3, lat,  4096,  128,  256);
  gemm_lrelu_kernel<<<dim3(2, 64), blk, 0, stream>>>(lat, Wd1, bd1, dd1,  4096,  256,  128);
  gemm_lrelu_kernel<<<dim3(4, 64), blk, 0, stream>>>(dd1, Wd2, bd2, dd2,  4096,  512,  256);
  gemm_lrelu_kernel<<<dim3(8, 64), blk, 0, stream>>>(dd2, Wd3, bd3, pred, 4096, 1024,  512);

  row_sumsq_kernel<<<4096, blk, 0, stream>>>(x,   rxv, 1024);
  row_sumsq_kernel<<<4096, blk, 0, stream>>>(lat, rlv,  128);

  dist_stats_kernel<<<dim3(32, 32), blk, 0, stream>>>(x, lat, rxv, rlv, partials);
  reduce_corr_kernel<<<1, blk, 0, stream>>>(partials, 1024, corr);
}